// Generator_13915694039527
// MI455X (gfx1250) — compile-verified
//
#include <hip/hip_runtime.h>
#include <hip/hip_bf16.h>
#include <math.h>

#define NPTS   1024
#define BATCH  4
#define KNN    16
#define M4     (4 * NPTS)

typedef float v2f __attribute__((ext_vector_type(2)));
typedef float v8f __attribute__((ext_vector_type(8)));

typedef __attribute__((address_space(1))) int* gint_p;   // global (AS1) int*
typedef __attribute__((address_space(3))) int* lint_p;   // LDS (AS3) int*

#if defined(__has_builtin)
#if __has_builtin(__builtin_amdgcn_wmma_f32_16x16x4_f32)
#define HAVE_WMMA_F32X4 1
#endif
#if __has_builtin(__builtin_amdgcn_global_load_async_to_lds_b32) && \
    __has_builtin(__builtin_amdgcn_s_wait_asynccnt)
#define HAVE_ASYNC_LDS 1
#endif
#endif

#define WMMA4(A, B, C) __builtin_amdgcn_wmma_f32_16x16x4_f32( \
    false, (A), false, (B), (short)0, (C), false, false)

// ---------------------------------------------------------------------------
// Generic fp32 WMMA GEMM:  Y[b] = act(W * X[b] + bias)
// W: [Co, Ci] row-major.  X: per-batch [Ci, P] row stride P.  REQUIRES P % 64 == 0.
// One wave computes a 16(M) x 64(N) tile: 4 accumulators sharing one A fragment
// per k-step -> 4 independent v_wmma_f32_16x16x4_f32 per A load, branch-free
// main loop (A row clamped; K-tail handled separately with B zeroing).
// ---------------------------------------------------------------------------
__global__ __launch_bounds__(32)
void wmma_gemm_f32(const float* __restrict__ W, const float* __restrict__ bias,
                   const float* __restrict__ X, float* __restrict__ Y,
                   int Co, int Ci, int P, long xBatch, long yBatch, int relu)
{
  const int b    = blockIdx.z;
  const float* Xb = X + (long)b * xBatch;
  float*       Yb = Y + (long)b * yBatch;
  const int m0   = blockIdx.x * 16;
  const int n0   = blockIdx.y * 64;
  const int lane = threadIdx.x;      // 0..31
  const int half = lane >> 4;        // 0/1
  const int lm   = lane & 15;

  const int rowA     = (m0 + lm < Co) ? (m0 + lm) : (Co - 1);  // clamped: no branch
  const float* Wr    = W + (long)rowA * Ci;
  const float* Xc    = Xb + n0 + lm;                           // per-lane column base

  v8f acc0 = {}, acc1 = {}, acc2 = {}, acc3 = {};

#ifdef HAVE_WMMA_F32X4
  const int CiMain = Ci & ~3;
  int k = 0;
  for (; k < CiMain; k += 4) {
    const int ka = k + 2 * half;
    v2f a;
    a.x = Wr[ka];                     // contiguous pair -> global_load_b64
    a.y = Wr[ka + 1];
    const float* x0 = Xc + (long)ka * P;
    const float* x1 = Xc + (long)(ka + 1) * P;
    v2f b0, b1, b2, b3;
    b0.x = x0[0];  b0.y = x1[0];
    b1.x = x0[16]; b1.y = x1[16];
    b2.x = x0[32]; b2.y = x1[32];
    b3.x = x0[48]; b3.y = x1[48];
    acc0 = WMMA4(a, b0, acc0);
    acc1 = WMMA4(a, b1, acc1);
    acc2 = WMMA4(a, b2, acc2);
    acc3 = WMMA4(a, b3, acc3);
  }
  if (k < Ci) {                       // tail (Ci % 4 != 0): zero B for invalid k
    const int ka  = k + 2 * half;
    const int k0  = (ka     < Ci) ? ka     : (Ci - 1);
    const int k1  = (ka + 1 < Ci) ? ka + 1 : (Ci - 1);
    const bool v0 = (ka     < Ci);
    const bool v1 = (ka + 1 < Ci);
    v2f a;
    a.x = Wr[k0];
    a.y = Wr[k1];
    const float* x0 = Xc + (long)k0 * P;
    const float* x1 = Xc + (long)k1 * P;
    v2f b0, b1, b2, b3;
    b0.x = v0 ? x0[0]  : 0.0f;  b0.y = v1 ? x1[0]  : 0.0f;
    b1.x = v0 ? x0[16] : 0.0f;  b1.y = v1 ? x1[16] : 0.0f;
    b2.x = v0 ? x0[32] : 0.0f;  b2.y = v1 ? x1[32] : 0.0f;
    b3.x = v0 ? x0[48] : 0.0f;  b3.y = v1 ? x1[48] : 0.0f;
    acc0 = WMMA4(a, b0, acc0);
    acc1 = WMMA4(a, b1, acc1);
    acc2 = WMMA4(a, b2, acc2);
    acc3 = WMMA4(a, b3, acc3);
  }
#else
  for (int kk = 0; kk < Ci; ++kk) {
    for (int v = 0; v < 8; ++v) {
      const int row = m0 + v + 8 * half;
      if (row < Co) {
        const float w = W[(long)row * Ci + kk];
        acc0[v] += w * Xc[(long)kk * P + 0];
        acc1[v] += w * Xc[(long)kk * P + 16];
        acc2[v] += w * Xc[(long)kk * P + 32];
        acc3[v] += w * Xc[(long)kk * P + 48];
      }
    }
  }
#endif

  const int colBase = n0 + lm;
  for (int v = 0; v < 8; ++v) {
    const int row = m0 + v + 8 * half;
    if (row < Co) {
      const float bv = bias ? bias[row] : 0.0f;
      float f0 = acc0[v] + bv, f1 = acc1[v] + bv,
            f2 = acc2[v] + bv, f3 = acc3[v] + bv;
      if (relu) {
        f0 = fmaxf(f0, 0.0f); f1 = fmaxf(f1, 0.0f);
        f2 = fmaxf(f2, 0.0f); f3 = fmaxf(f3, 0.0f);
      }
      float* yr = Yb + (long)row * P + colBase;
      yr[0] = f0; yr[16] = f1; yr[32] = f2; yr[48] = f3;
    }
  }
}

// ---------------------------------------------------------------------------
// kNN, LDS-staged: each block stages the whole per-batch [C, N] feature matrix
// into LDS (<= 192KB, fits CDNA5's 320KB WGP LDS) via ASYNC copies, then each
// thread scans all m from LDS (same-m reads broadcast across the wave).
// Keeps the 17 smallest squared distances (stable ties like lax.top_k),
// drops self, writes idx[b,k,n].  C is a template param so the per-point
// feature vector lives in registers and the distance loop fully unrolls.
// ---------------------------------------------------------------------------
template <int C>
__global__ __launch_bounds__(256)
void k_knn(const float* __restrict__ x, int* __restrict__ idx)
{
  extern __shared__ float sh[];                      // [C * NPTS]
  const int chunk = blockIdx.x % (NPTS / 256);
  const int b     = blockIdx.x / (NPTS / 256);
  const float* xb = x + (long)b * C * NPTS;

  // cooperative stage of xb -> LDS
  for (int i = threadIdx.x; i < C * NPTS; i += 256) {
#ifdef HAVE_ASYNC_LDS
    __builtin_amdgcn_global_load_async_to_lds_b32(
        (gint_p)(xb + i), (lint_p)(sh + i), 0, 0);
#else
    sh[i] = xb[i];
#endif
  }
#ifdef HAVE_ASYNC_LDS
  __builtin_amdgcn_s_wait_asynccnt(0);
#endif
  __syncthreads();

  const int n = chunk * 256 + threadIdx.x;

  float xn[C];
#pragma unroll
  for (int c = 0; c < C; ++c) xn[c] = sh[c * NPTS + n];

  float bd[KNN + 1];
  int   bi[KNN + 1];
  for (int i = 0; i <= KNN; ++i) { bd[i] = 3.0e38f; bi[i] = 0; }

  for (int m = 0; m < NPTS; ++m) {
    float d = 0.0f;
#pragma unroll
    for (int c = 0; c < C; ++c) {
      const float df = xn[c] - sh[c * NPTS + m];
      d += df * df;
    }
    if (d < bd[KNN]) {
      int j = KNN;
      while (j > 0 && d < bd[j - 1]) { bd[j] = bd[j - 1]; bi[j] = bi[j - 1]; --j; }
      bd[j] = d; bi[j] = m;
    }
  }
  int* ib = idx + (long)b * KNN * NPTS;
  for (int k = 0; k < KNN; ++k) ib[(long)k * NPTS + n] = bi[k + 1];
}

// Edge features into scratch T[B, Tctot, K*N] at channel offset chOff:
//   ch[chOff+c]     = central,  ch[chOff+C+c] = neighbor - central
__global__ void k_edge(const float* __restrict__ pc, const int* __restrict__ idx,
                       int C, float* __restrict__ T, int chOff, int Tctot, long total)
{
  long t = (long)blockIdx.x * blockDim.x + threadIdx.x;
  if (t >= total) return;
  const int n = (int)(t % NPTS);
  const int k = (int)((t / NPTS) % KNN);
  const int c = (int)((t / ((long)NPTS * KNN)) % C);
  const int b = (int)(t / ((long)NPTS * KNN * C));
  const float ctr = pc[((long)b * C + c) * NPTS + n];
  const int   nb  = idx[((long)b * KNN + k) * NPTS + n];
  const float nv  = pc[((long)b * C + c) * NPTS + nb];
  float* Tb = T + (long)b * Tctot * (KNN * NPTS);
  const long pos = (long)k * NPTS + n;
  Tb[(long)(chOff + c)     * (KNN * NPTS) + pos] = ctr;
  Tb[(long)(chOff + C + c) * (KNN * NPTS) + pos] = nv - ctr;
}

// out[b, c, n] = max_k T[b, c, k*N + n]   (c < Ct), out buffer has outCtot chans
__global__ void k_maxk(const float* __restrict__ T, float* __restrict__ out,
                       int Ct, int Tctot, int outCtot)
{
  const long t = (long)blockIdx.x * blockDim.x + threadIdx.x;
  if (t >= (long)BATCH * Ct * NPTS) return;
  const int n = (int)(t % NPTS);
  const int c = (int)((t / NPTS) % Ct);
  const int b = (int)(t / ((long)NPTS * Ct));
  const float* Tb = T + ((long)b * Tctot + c) * (KNN * NPTS);
  float mx = Tb[n];
  for (int k = 1; k < KNN; ++k) mx = fmaxf(mx, Tb[(long)k * NPTS + n]);
  out[((long)b * outCtot + c) * NPTS + n] = mx;
}

// dst[b, dstOff+c, p] = src[b, srcOff+c, p]
__global__ void k_copych(const float* __restrict__ src, float* __restrict__ dst,
                         int srcOff, int srcCtot, int dstOff, int dstCtot, int C, int P)
{
  const long t = (long)blockIdx.x * blockDim.x + threadIdx.x;
  if (t >= (long)BATCH * C * P) return;
  const int p = (int)(t % P);
  const int c = (int)((t / P) % C);
  const int b = (int)(t / ((long)P * C));
  dst[((long)b * dstCtot + dstOff + c) * P + p] =
      src[((long)b * srcCtot + srcOff + c) * P + p];
}

// net[b, c, j]: c<128 -> x[b,c,j%N]; c==128/129 -> GRID[j/N]
__global__ void k_net_build(const float* __restrict__ x, float* __restrict__ net)
{
  const long t = (long)blockIdx.x * blockDim.x + threadIdx.x;
  if (t >= (long)BATCH * 130 * M4) return;
  const int j = (int)(t % M4);
  const int c = (int)((t / M4) % 130);
  const int b = (int)(t / ((long)M4 * 130));
  const float gx[4] = {-0.2f, -0.2f, 0.2f, 0.2f};
  const float gy[4] = {-0.2f,  0.2f, -0.2f, 0.2f};
  float v;
  if (c < 128)      v = x[((long)b * 128 + c) * NPTS + (j % NPTS)];
  else if (c == 128) v = gx[j / NPTS];
  else               v = gy[j / NPTS];
  net[t] = v;
}

__global__ void k_transpose(const float* __restrict__ src, float* __restrict__ dst,
                            int C, int P)
{
  const long t = (long)blockIdx.x * blockDim.x + threadIdx.x;
  if (t >= (long)C * P) return;
  const int p = (int)(t % P), c = (int)(t / P);
  dst[(long)p * C + c] = src[(long)c * P + p];
}

__global__ void k_softmax_rows(float* __restrict__ S, int M)
{
  const int row = blockIdx.x;
  float* r = S + (long)row * M;
  __shared__ float red[256];
  float mx = -3.0e38f;
  for (int j = threadIdx.x; j < M; j += 256) mx = fmaxf(mx, r[j]);
  red[threadIdx.x] = mx; __syncthreads();
  for (int s = 128; s > 0; s >>= 1) {
    if (threadIdx.x < s) red[threadIdx.x] = fmaxf(red[threadIdx.x], red[threadIdx.x + s]);
    __syncthreads();
  }
  mx = red[0]; __syncthreads();
  float sum = 0.0f;
  for (int j = threadIdx.x; j < M; j += 256) {
    const float e = expf(r[j] - mx); r[j] = e; sum += e;
  }
  red[threadIdx.x] = sum; __syncthreads();
  for (int s = 128; s > 0; s >>= 1) {
    if (threadIdx.x < s) red[threadIdx.x] += red[threadIdx.x + s];
    __syncthreads();
  }
  const float inv = 1.0f / red[0];
  for (int j = threadIdx.x; j < M; j += 256) r[j] *= inv;
}

__global__ void k_resid(const float* __restrict__ o, const float* __restrict__ x,
                        const float* __restrict__ gamma, float* __restrict__ out, long total)
{
  const long t = (long)blockIdx.x * blockDim.x + threadIdx.x;
  if (t < total) out[t] = gamma[0] * o[t] + x[t];
}

__global__ void k_sub(const float* a, const float* b, float* o, long total)
{
  const long t = (long)blockIdx.x * blockDim.x + threadIdx.x;
  if (t < total) o[t] = a[t] - b[t];
}

__global__ void k_add(const float* a, const float* b, float* o, long total)
{
  const long t = (long)blockIdx.x * blockDim.x + threadIdx.x;
  if (t < total) o[t] = a[t] + b[t];
}

// xp[b, i*4+r, n] = H[b, i, r*N + n]   ([B,128,4N] -> [B,512,N])
__global__ void k_pack_down(const float* __restrict__ H, float* __restrict__ xp)
{
  const long t = (long)blockIdx.x * blockDim.x + threadIdx.x;
  if (t >= (long)BATCH * 512 * NPTS) return;
  const int n  = (int)(t % NPTS);
  const int kk = (int)((t / NPTS) % 512);
  const int b  = (int)(t / ((long)NPTS * 512));
  const int i = kk >> 2, r = kk & 3;
  xp[t] = H[((long)b * 128 + i) * M4 + (long)r * NPTS + n];
}

// ---------------------------------------------------------------------------
// Host-side orchestration
// ---------------------------------------------------------------------------
static inline void gemm(hipStream_t s, const float* W, const float* bias,
                        const float* X, float* Y, int Co, int Ci, int P,
                        long xB, long yB, int batches, int relu)
{
  dim3 g((Co + 15) / 16, (unsigned)(P / 64), batches);   // P % 64 == 0 at all call sites
  wmma_gemm_f32<<<g, dim3(32), 0, s>>>(W, bias, X, Y, Co, Ci, P, xB, yB, relu);
}

#define L1D(kern, total, ...) \
  kern<<<dim3((unsigned)(((total) + 255) / 256)), dim3(256), 0, stream>>>(__VA_ARGS__)

struct DnP { const float *w1, *b1, *w2, *b2, *w3, *b3; };
struct UpP { const float *wF, *bF, *wG, *bG, *wH, *bH, *gamma, *w1, *b1, *w2, *b2; };

static void dense_conv(hipStream_t stream, const float* pcL, int C, const DnP& p,
                       int* idxBuf, float* T, float* outBuf, int outCtot)
{
  const int P = KNN * NPTS, CT = 168;
  const long cb = (long)CT * P;
  {
    dim3 g(BATCH * (NPTS / 256));
    const size_t shb = (size_t)C * NPTS * sizeof(float);   // 96KB / 192KB, < 320KB WGP LDS
    if (C == 24)      k_knn<24><<<g, dim3(256), shb, stream>>>(pcL, idxBuf);
    else /* C == 48*/ k_knn<48><<<g, dim3(256), shb, stream>>>(pcL, idxBuf);
  }
  {
    long tot = (long)BATCH * C * KNN * NPTS;
    L1D(k_edge, tot, pcL, idxBuf, C, T, 72, CT, tot);
  }
  gemm(stream, p.w1, p.b1, T + 72L * P, T + 48L * P, 24, 2 * C,      P, cb, cb, BATCH, 1);
  gemm(stream, p.w2, p.b2, T + 48L * P, T + 24L * P, 24, 24 + 2 * C, P, cb, cb, BATCH, 1);
  gemm(stream, p.w3, p.b3, T + 24L * P, T,           24, 48 + 2 * C, P, cb, cb, BATCH, 0);
  const int Ct = 72 + 2 * C;
  L1D(k_maxk, (long)BATCH * Ct * NPTS, T, outBuf, Ct, CT, outCtot);
}

static void run_up_block(hipStream_t stream, const float* x, const UpP& p,
                         float* net, float* Fb, float* Gb, float* Hb, float* gT,
                         float* S, float* Ob, float* Att, float* Hid, float* Hout)
{
  const int C = 130, Cq = 32, M = M4;
  const long tot = (long)BATCH * C * M;
  L1D(k_net_build, tot, x, net);
  gemm(stream, p.wF, p.bF, net, Fb, Cq, C, M, (long)C * M, (long)Cq * M, BATCH, 1);
  gemm(stream, p.wG, p.bG, net, Gb, Cq, C, M, (long)C * M, (long)Cq * M, BATCH, 1);
  gemm(stream, p.wH, p.bH, net, Hb, C,  C, M, (long)C * M, (long)C  * M, BATCH, 1);
  for (int b = 0; b < BATCH; ++b) {
    const float* fb = Fb + (long)b * Cq * M;
    const float* gb = Gb + (long)b * Cq * M;
    const float* hb = Hb + (long)b * C  * M;
    L1D(k_transpose, (long)Cq * M, gb, gT, Cq, M);
    gemm(stream, gT, nullptr, fb, S, M, Cq, M, 0, 0, 1, 0);
    k_softmax_rows<<<dim3(M), dim3(256), 0, stream>>>(S, M);
    gemm(stream, hb, nullptr, S, Ob + (long)b * C * M, C, M, M, 0, 0, 1, 0);
  }
  L1D(k_resid, tot, Ob, net, p.gamma, Att, tot);
  gemm(stream, p.w1, p.b1, Att, Hid, 256, C,   M, (long)C * M,   256L * M, BATCH, 1);
  gemm(stream, p.w2, p.b2, Hid, Hout, 128, 256, M, 256L * M,     128L * M, BATCH, 1);
}

extern "C" void kernel_launch(void* const* d_in, const int* in_sizes, int n_in,
                              void* d_out, int out_size, void* d_ws, size_t ws_size,
                              hipStream_t stream)
{
  (void)in_sizes; (void)n_in; (void)out_size; (void)ws_size;
  int ii = 0;
  auto in = [&](void) { return (const float*)d_in[ii++]; };

  const float* pc   = in();
  const float* fe1w = in(); const float* fe1b = in();
  DnP d1; d1.w1=in(); d1.b1=in(); d1.w2=in(); d1.b2=in(); d1.w3=in(); d1.b3=in();
  const float* fe2w = in(); const float* fe2b = in();
  DnP d2; d2.w1=in(); d2.b1=in(); d2.w2=in(); d2.b2=in(); d2.w3=in(); d2.b3=in();
  const float* fe3w = in(); const float* fe3b = in();
  DnP d3; d3.w1=in(); d3.b1=in(); d3.w2=in(); d3.b2=in(); d3.w3=in(); d3.b3=in();
  const float* fe4w = in(); const float* fe4b = in();
  DnP d4; d4.w1=in(); d4.b1=in(); d4.w2=in(); d4.b2=in(); d4.w3=in(); d4.b3=in();
  const float* upw  = in(); const float* upb  = in();
  UpP u1; u1.wF=in(); u1.bF=in(); u1.wG=in(); u1.bG=in(); u1.wH=in(); u1.bH=in();
          u1.gamma=in(); u1.w1=in(); u1.b1=in(); u1.w2=in(); u1.b2=in();
  UpP u2; u2.wF=in(); u2.bF=in(); u2.wG=in(); u2.bG=in(); u2.wH=in(); u2.bH=in();
          u2.gamma=in(); u2.w1=in(); u2.b1=in(); u2.w2=in(); u2.b2=in();
  const float* dbw1 = in(); const float* dbb1 = in();
  const float* dbw2 = in(); const float* dbb2 = in();
  const float* g1w  = in(); const float* g1b  = in();
  const float* g2w  = in(); const float* g2b  = in();

  // ---- workspace bump allocator (256B aligned) ----
  char* wp = (char*)d_ws;
  auto allocF = [&](long nflt) {
    float* r = (float*)wp;
    wp += ((nflt * 4 + 255) / 256) * 256;
    return r;
  };
  int*   idxB  = (int*)allocF((long)BATCH * KNN * NPTS);
  // Union region: EdgeConv scratch T (11.0M f) vs attention S+gT (16.9M f)
  float* R0    = allocF(16777216L + 131072L);
  float* T     = R0;                 // [B,168,K*N]
  float* S     = R0;                 // [M,M]
  float* gT    = R0 + 16777216L;     // [M,32]
  float* L0f   = allocF((long)BATCH * 24  * NPTS);
  float* L1    = allocF((long)BATCH * 144 * NPTS);
  float* Tmp48 = allocF((long)BATCH * 48  * NPTS);
  float* L2    = allocF((long)BATCH * 312 * NPTS);
  float* L3    = allocF((long)BATCH * 480 * NPTS);
  float* Feats = allocF((long)BATCH * 648 * NPTS);
  float* Lbuf  = allocF((long)BATCH * 128 * NPTS);
  float* net   = allocF((long)BATCH * 130 * M4);
  float* Fb    = allocF((long)BATCH * 32  * M4);
  float* Gb    = allocF((long)BATCH * 32  * M4);
  float* Hb    = allocF((long)BATCH * 130 * M4);
  float* Ob    = allocF((long)BATCH * 130 * M4);
  float* Att   = allocF((long)BATCH * 130 * M4);
  float* Hid   = allocF((long)BATCH * 256 * M4);
  float* H0    = allocF((long)BATCH * 128 * M4);
  float* H1    = allocF((long)BATCH * 128 * M4);
  float* H2    = allocF((long)BATCH * 128 * M4);
  float* X2    = allocF((long)BATCH * 128 * NPTS);
  float* xp    = allocF((long)BATCH * 512 * NPTS);
  float* dh    = allocF((long)BATCH * 256 * NPTS);
  float* L0d   = allocF((long)BATCH * 128 * NPTS);
  float* cH    = allocF((long)BATCH * 64  * M4);

  // ---- feature extraction ----
  gemm(stream, fe1w, fe1b, pc, L0f, 24, 3, NPTS, 3L * NPTS, 24L * NPTS, BATCH, 1);

  dense_conv(stream, L0f, 24, d1, idxB, T, L1, 144);
  L1D(k_copych, (long)BATCH * 24 * NPTS, L0f, L1, 0, 24, 120, 144, 24, NPTS);
  gemm(stream, fe2w, fe2b, L1, Tmp48, 48, 144, NPTS, 144L * NPTS, 48L * NPTS, BATCH, 1);

  dense_conv(stream, Tmp48, 48, d2, idxB, T, L2, 312);
  L1D(k_copych, (long)BATCH * 144 * NPTS, L1, L2, 0, 144, 168, 312, 144, NPTS);
  gemm(stream, fe3w, fe3b, L2, Tmp48, 48, 312, NPTS, 312L * NPTS, 48L * NPTS, BATCH, 1);

  dense_conv(stream, Tmp48, 48, d3, idxB, T, L3, 480);
  L1D(k_copych, (long)BATCH * 312 * NPTS, L2, L3, 0, 312, 168, 480, 312, NPTS);
  gemm(stream, fe4w, fe4b, L3, Tmp48, 48, 480, NPTS, 480L * NPTS, 48L * NPTS, BATCH, 1);

  dense_conv(stream, Tmp48, 48, d4, idxB, T, Feats, 648);
  L1D(k_copych, (long)BATCH * 480 * NPTS, L3, Feats, 0, 480, 168, 648, 480, NPTS);

  gemm(stream, upw, upb, Feats, Lbuf, 128, 648, NPTS, 648L * NPTS, 128L * NPTS, BATCH, 1);

  // ---- up / down / up ----
  run_up_block(stream, Lbuf, u1, net, Fb, Gb, Hb, gT, S, Ob, Att, Hid, H0);

  L1D(k_pack_down, (long)BATCH * 512 * NPTS, H0, xp);
  gemm(stream, dbw1, dbb1, xp, dh, 256, 512, NPTS, 512L * NPTS, 256L * NPTS, BATCH, 1);
  gemm(stream, dbw2, dbb2, dh, L0d, 128, 256, NPTS, 256L * NPTS, 128L * NPTS, BATCH, 1);

  L1D(k_sub, (long)BATCH * 128 * NPTS, L0d, Lbuf, X2, (long)BATCH * 128 * NPTS);
  run_up_block(stream, X2, u2, net, Fb, Gb, Hb, gT, S, Ob, Att, Hid, H1);

  L1D(k_add, (long)BATCH * 128 * M4, H0, H1, H2, (long)BATCH * 128 * M4);

  // ---- coordinate head ----
  gemm(stream, g1w, g1b, H2, cH, 64, 128, M4, 128L * M4, 64L * M4, BATCH, 1);
  gemm(stream, g2w, g2b, cH, (float*)d_out, 3, 64, M4, 64L * M4, 3L * M4, BATCH, 0);
}